// HungarianMatcher_22600117911557
// MI455X (gfx1250) — compile-verified
//
#include <hip/hip_runtime.h>

// ---------------------------------------------------------------------------
// HungarianMatcher fused cost matrix for gfx1250 (MI455X).
//   C[q,t] = L1(pbox_q, tbox_t) + focal[q, cls_t] - GIoU(q,t)
// Sizes fixed by the reference.
// ---------------------------------------------------------------------------
#define BS   32
#define NQ   900
#define NC   91
#define NT   1600
#define QTOT (BS * NQ)   // 28800
#define NCP  92          // classes padded to even (col 91 == 0)
#define PSTR 12          // per-box aux stride (floats)

typedef __attribute__((ext_vector_type(2))) float v2f;
typedef __attribute__((ext_vector_type(8))) float v8f;

#define HAVE_WMMA_F32X4 __has_builtin(__builtin_amdgcn_wmma_f32_16x16x4_f32)
#define HAVE_ASYNC_LDS  __has_builtin(__builtin_amdgcn_global_load_async_to_lds_b32)
#define HAVE_WAIT_ASYNC __has_builtin(__builtin_amdgcn_s_wait_asynccnt)

// ---------------------------------------------------------------------------
// Kernel 1: focal-style classification cost table, focal[q][c], padded to 92.
//   pos = a*(1-p)^2 * -log(p+eps);  neg = (1-a)*p^2 * -log(1-p+eps)
// ---------------------------------------------------------------------------
__global__ __launch_bounds__(256)
void focal_kernel(const float* __restrict__ logits, float* __restrict__ focal) {
    int idx = blockIdx.x * 256 + threadIdx.x;
    if (idx >= QTOT * NCP) return;
    int q = idx / NCP;
    int c = idx - q * NCP;
    float f = 0.0f;
    if (c < NC) {
        float x   = logits[q * NC + c];
        float p   = 1.0f / (1.0f + __expf(-x));
        float omp = 1.0f - p;
        float pos = 0.25f * omp * omp * (-__logf(p   + 1e-8f));
        float neg = 0.75f * p   * p   * (-__logf(omp + 1e-8f));
        f = pos - neg;
    }
    focal[idx] = f;
}

// ---------------------------------------------------------------------------
// Kernel 2: per-box aux: [cx,cy,w,h, x1,y1,x2,y2, area, cls, 0,0]
// ---------------------------------------------------------------------------
__global__ __launch_bounds__(256)
void boxaux_kernel(const float* __restrict__ boxes, const int* __restrict__ ids,
                   float* __restrict__ aux, int n) {
    int i = blockIdx.x * 256 + threadIdx.x;
    if (i >= n) return;
    float cx = boxes[i * 4 + 0], cy = boxes[i * 4 + 1];
    float w  = boxes[i * 4 + 2], h  = boxes[i * 4 + 3];
    float x1 = cx - 0.5f * w, y1 = cy - 0.5f * h;
    float x2 = cx + 0.5f * w, y2 = cy + 0.5f * h;
    float* o = aux + (size_t)i * PSTR;
    o[0] = cx; o[1] = cy; o[2] = w;  o[3] = h;
    o[4] = x1; o[5] = y1; o[6] = x2; o[7] = y2;
    o[8] = (x2 - x1) * (y2 - y1);
    o[9] = ids ? __int_as_float(ids[i]) : 0.0f;
    o[10] = 0.0f; o[11] = 0.0f;
}

// ---------------------------------------------------------------------------
// Async global->LDS staging (CDNA5 GLOBAL_LOAD_ASYNC_TO_LDS), with fallback.
// Builtin param 0 is a global-AS int* per hipcc diagnostic; cast through
// integers (AS1 ptrs are 64-bit; AS3 ptr = low 32 bits of generic LDS addr).
// ---------------------------------------------------------------------------
#if HAVE_ASYNC_LDS
__device__ __forceinline__ void cp_async_f32(float* l, const float* g) {
    typedef __attribute__((address_space(1))) int GI;
    typedef __attribute__((address_space(3))) int SI;
    __builtin_amdgcn_global_load_async_to_lds_b32(
        (GI*)(unsigned long long)(size_t)g,
        (SI*)(unsigned int)(size_t)l, 0, 0);
}
#endif

__device__ __forceinline__ void stage(float* dst, const float* src, int n, int tid) {
#if HAVE_ASYNC_LDS
    for (int i = tid; i < n; i += 256) cp_async_f32(dst + i, src + i);
#else
    for (int i = tid; i < n; i += 256) dst[i] = src[i];
#endif
}

__device__ __forceinline__ void wait_async0() {
#if HAVE_ASYNC_LDS
#if HAVE_WAIT_ASYNC
    __builtin_amdgcn_s_wait_asynccnt(0);
#else
    asm volatile("s_wait_asynccnt 0" ::: "memory");
#endif
#endif
}

// ---------------------------------------------------------------------------
// Kernel 3: pairwise tile kernel. Block = 256 thr = 8 waves.
// Block tile: 32 q x 64 t. Wave tile: 16 q x 16 t.
// cost_class via exact one-hot f32 WMMA (focal[16x92] x onehot[92x16]).
// ---------------------------------------------------------------------------
__global__ __launch_bounds__(256)
void cost_tile_kernel(const float* __restrict__ wsF, const float* __restrict__ wsP,
                      const float* __restrict__ wsT, float* __restrict__ out) {
    __shared__ float sF[32 * NCP];   // focal rows for this block's 32 queries
    __shared__ float sQ[32 * PSTR];  // pred aux
    __shared__ float sT[64 * PSTR];  // tgt aux

    const int tid = threadIdx.x;
    const int bt  = blockIdx.x;      // 0..24  (64 targets each)
    const int bq  = blockIdx.y;      // 0..899 (32 queries each)

    stage(sF, wsF + (size_t)bq * (32 * NCP),  32 * NCP,  tid);
    stage(sQ, wsP + (size_t)bq * (32 * PSTR), 32 * PSTR, tid);
    stage(sT, wsT + (size_t)bt * (64 * PSTR), 64 * PSTR, tid);
    wait_async0();
    __syncthreads();

    const int lane = tid & 31;
    const int wv   = tid >> 5;       // wave 0..7
    const int li   = lane & 15;      // sub-lane 0..15
    const int h    = lane >> 4;      // half 0/1
    const int qw   = wv >> 2;        // q-tile of wave: 0..1
    const int tw   = wv & 3;         // t-tile of wave: 0..3

    const int tl  = tw * 16 + li;    // target within block tile
    const int cls = __float_as_int(sT[tl * PSTR + 9]);

#if HAVE_WMMA_F32X4
    // A (16x4 f32): lanes 0-15 hold row M=li with {K=k0,k0+1}; lanes 16-31 {k0+2,k0+3}.
    // B (4x16 f32): VGPR0 = row K=k0 (lanes<16) / K=k0+2 (lanes>=16); VGPR1 = +1.
    // D (16x16 f32): VGPR v -> M = v + 8*h, N = li.
    v8f acc = {0.f, 0.f, 0.f, 0.f, 0.f, 0.f, 0.f, 0.f};
    const int arow = qw * 16 + li;
#pragma unroll
    for (int k0 = 0; k0 < NCP; k0 += 4) {
        const int ka = k0 + 2 * h;
        v2f a = *(const v2f*)&sF[arow * NCP + ka];  // 8B aligned: row stride 92, ka even
        v2f b;
        b.x = (cls == ka    ) ? 1.0f : 0.0f;
        b.y = (cls == ka + 1) ? 1.0f : 0.0f;
        acc = __builtin_amdgcn_wmma_f32_16x16x4_f32(
            false, a, false, b, (short)0, acc, false, false);
    }
#endif

    // Per-lane epilogue: fixed target, 8 query rows (D-matrix layout).
    const float* T = &sT[tl * PSTR];
    const float tcx = T[0], tcy = T[1], twd = T[2], tht = T[3];
    const float bx1 = T[4], by1 = T[5], bx2 = T[6], by2 = T[7], areaB = T[8];

    const int    qbase = qw * 16 + 8 * h;
    const size_t gq0   = (size_t)bq * 32 + qbase;
    const int    tg    = bt * 64 + tl;

#pragma unroll
    for (int v = 0; v < 8; ++v) {
        const float* Qd = &sQ[(qbase + v) * PSTR];
        const float pcx = Qd[0], pcy = Qd[1], pwd = Qd[2], pht = Qd[3];
        const float ax1 = Qd[4], ay1 = Qd[5], ax2 = Qd[6], ay2 = Qd[7], areaA = Qd[8];

        // L1 box cost
        float cb = fabsf(pcx - tcx) + fabsf(pcy - tcy)
                 + fabsf(pwd - twd) + fabsf(pht - tht);

        // GIoU
        float iw    = fmaxf(fminf(ax2, bx2) - fmaxf(ax1, bx1), 0.0f);
        float ih    = fmaxf(fminf(ay2, by2) - fmaxf(ay1, by1), 0.0f);
        float inter = iw * ih;
        float uni   = areaA + areaB - inter;
        float ew    = fmaxf(ax2, bx2) - fminf(ax1, bx1);
        float eh    = fmaxf(ay2, by2) - fminf(ay1, by1);
        float enc   = ew * eh;
        float giou  = __fdividef(inter, uni) - __fdividef(enc - uni, enc);

#if HAVE_WMMA_F32X4
        float cc = acc[v];
#else
        float cc = sF[(qbase + v) * NCP + cls];
#endif
        out[(gq0 + v) * (size_t)NT + tg] = cb + cc - giou;
    }
}

// ---------------------------------------------------------------------------
// Launch
// ---------------------------------------------------------------------------
extern "C" void kernel_launch(void* const* d_in, const int* in_sizes, int n_in,
                              void* d_out, int out_size, void* d_ws, size_t ws_size,
                              hipStream_t stream) {
    const float* logits = (const float*)d_in[0];  // [32,900,91]
    const float* pboxes = (const float*)d_in[1];  // [32,900,4]
    const float* tboxes = (const float*)d_in[2];  // [1600,4]
    const int*   tids   = (const int*)  d_in[3];  // [1600]
    float* out = (float*)d_out;                   // [32,900,1600]

    float* wsF = (float*)d_ws;                         // QTOT*NCP  (~10.6 MB)
    float* wsP = wsF + (size_t)QTOT * NCP;             // QTOT*PSTR (~1.4 MB)
    float* wsT = wsP + (size_t)QTOT * PSTR;            // NT*PSTR   (~77 KB)

    focal_kernel<<<(QTOT * NCP + 255) / 256, 256, 0, stream>>>(logits, wsF);
    boxaux_kernel<<<(QTOT + 255) / 256, 256, 0, stream>>>(pboxes, nullptr, wsP, QTOT);
    boxaux_kernel<<<(NT + 255) / 256, 256, 0, stream>>>(tboxes, tids, wsT, NT);

    dim3 grid(NT / 64, QTOT / 32);  // 25 x 900, exact
    cost_tile_kernel<<<grid, 256, 0, stream>>>(wsF, wsP, wsT, out);
}